// DiMap_2_34832184770680
// MI455X (gfx1250) — compile-verified
//
#include <hip/hip_runtime.h>

// ---------------------------------------------------------------------------
// SPDNet layer for MI455X (gfx1250, wave32, WMMA).
// Batched 64x64 SPD pipeline: pair barycenter -> SPD batchnorm -> ReEig.
// eigh = parallel cyclic Jacobi in LDS (1 workgroup / matrix), all
// reconstructions & congruences use V_WMMA_F32_16X16X4_F32 fp32 matrix ops.
// ---------------------------------------------------------------------------

#define NMAT   64
#define LDSP   66          // padded LDS row stride
#define NSWEEPS 10         // Jacobi sweeps (quadratic convergence, 64x64)

typedef __attribute__((ext_vector_type(2))) float v2f;
typedef __attribute__((ext_vector_type(8))) float v8f;

enum { M_SQRT_BOTH = 0, M_LOG = 1, M_EXP = 2, M_CLAMP = 3, M_SQRT = 4, M_INVSQRT = 5 };

// C(64x64) = A @ op(B). A,B in LDS (row-major, stride LDSP).
// transB: B(k,n) read as sB[n*LDSP+k] (i.e. op(B)=B^T of the stored array).
// dst: global or LDS, row stride dstStride. 256 threads = 8 waves, 2 tiles/wave.
__device__ __forceinline__ void gemm64(const float* __restrict__ sA,
                                       const float* __restrict__ sB,
                                       bool transB,
                                       float* __restrict__ dst, int dstStride)
{
    const int tid  = threadIdx.x;
    const int wave = tid >> 5;
    const int lane = tid & 31;
    const int half = lane >> 4;     // 0: K 0..1 | 1: K 2..3 (A frag layout)
    const int l16  = lane & 15;

    for (int t = wave; t < 16; t += 8) {
        const int mt = (t >> 2) << 4;   // M tile base
        const int nt = (t & 3)  << 4;   // N tile base
        v8f acc = {};
        for (int kb = 0; kb < NMAT; kb += 4) {
            const int kk = kb + (half << 1);
            v2f a, b;
            a.x = sA[(mt + l16) * LDSP + kk];
            a.y = sA[(mt + l16) * LDSP + kk + 1];
            if (transB) {
                b.x = sB[(nt + l16) * LDSP + kk];
                b.y = sB[(nt + l16) * LDSP + kk + 1];
            } else {
                b.x = sB[(kk)     * LDSP + (nt + l16)];
                b.y = sB[(kk + 1) * LDSP + (nt + l16)];
            }
            acc = __builtin_amdgcn_wmma_f32_16x16x4_f32(
                    false, a, false, b, (short)0, acc, false, false);
        }
        // D layout: VGPR r -> M = mt + r + 8*half, N = nt + l16
        const int row0 = mt + (half << 3);
        const int col  = nt + l16;
#pragma unroll
        for (int r = 0; r < 8; ++r)
            dst[(row0 + r) * dstStride + col] = acc[r];
    }
}

// ---------------------------------------------------------------------------
// Fused eigh + spectral-function reconstruction: out = V f(diag) V^T.
// One block (256 thr) per matrix; Jacobi on LDS copies of A and V.
// ---------------------------------------------------------------------------
template <int MODE>
__global__ __launch_bounds__(256)
void k_eigh_fn(const float* __restrict__ in, float* __restrict__ out1,
               float* __restrict__ out2)
{
    __shared__ float sA[NMAT * LDSP];
    __shared__ float sV[NMAT * LDSP];
    __shared__ float sEv[NMAT];
    __shared__ float sC[32], sS[32];
    __shared__ int   sPp[32], sQq[32];
    __shared__ int   top[32], bot[32];

    const int  tid  = threadIdx.x;
    const long base = (long)blockIdx.x * (NMAT * NMAT);

    for (int i = tid; i < NMAT * NMAT; i += 256) {
        const int r = i >> 6, c = i & 63;
        sA[r * LDSP + c] = in[base + i];
        sV[r * LDSP + c] = (r == c) ? 1.0f : 0.0f;
    }
    if (tid < 32) { top[tid] = 2 * tid; bot[tid] = 2 * tid + 1; }
    __syncthreads();

    for (int sweep = 0; sweep < NSWEEPS; ++sweep) {
        for (int rnd = 0; rnd < 63; ++rnd) {
            if (tid < 32) {                       // rotation angles
                int p = top[tid], q = bot[tid];
                if (p > q) { int t = p; p = q; q = t; }
                const float app = sA[p * LDSP + p];
                const float aqq = sA[q * LDSP + q];
                const float apq = sA[p * LDSP + q];
                float c = 1.0f, s = 0.0f;
                if (fabsf(apq) > 1e-30f) {
                    const float tau = (aqq - app) / (2.0f * apq);
                    const float t   = copysignf(1.0f, tau) /
                                      (fabsf(tau) + sqrtf(1.0f + tau * tau));
                    c = 1.0f / sqrtf(1.0f + t * t);
                    s = t * c;
                }
                sC[tid] = c; sS[tid] = s; sPp[tid] = p; sQq[tid] = q;
            }
            __syncthreads();
            {   // row phase: A <- J^T A   (pair = tid/8, 8 cols/thread)
                const int k = tid >> 3, j0 = (tid & 7) << 3;
                const int p = sPp[k], q = sQq[k];
                const float c = sC[k], s = sS[k];
#pragma unroll
                for (int jj = 0; jj < 8; ++jj) {
                    const int j = j0 + jj;
                    const float ap = sA[p * LDSP + j], aq = sA[q * LDSP + j];
                    sA[p * LDSP + j] = c * ap - s * aq;
                    sA[q * LDSP + j] = s * ap + c * aq;
                }
            }
            __syncthreads();
            {   // column phase: A <- A J ; V <- V J
                const int k = tid >> 3, i0 = (tid & 7) << 3;
                const int p = sPp[k], q = sQq[k];
                const float c = sC[k], s = sS[k];
#pragma unroll
                for (int ii = 0; ii < 8; ++ii) {
                    const int i = i0 + ii;
                    float ap = sA[i * LDSP + p], aq = sA[i * LDSP + q];
                    sA[i * LDSP + p] = c * ap - s * aq;
                    sA[i * LDSP + q] = s * ap + c * aq;
                    ap = sV[i * LDSP + p]; aq = sV[i * LDSP + q];
                    sV[i * LDSP + p] = c * ap - s * aq;
                    sV[i * LDSP + q] = s * ap + c * aq;
                }
            }
            __syncthreads();
            if (tid == 0) {                       // round-robin tournament
                const int nt1  = bot[0];
                const int last = top[31];
                for (int k = 31; k >= 2; --k) top[k] = top[k - 1];
                top[1] = nt1;
                for (int k = 0; k < 31; ++k) bot[k] = bot[k + 1];
                bot[31] = last;
            }
            __syncthreads();
        }
    }

    if (tid < NMAT) sEv[tid] = sA[tid * LDSP + tid];
    __syncthreads();

    // pass 1: scale V columns by f(lambda), GEMM (V.f) @ V^T
    for (int i = tid; i < NMAT * NMAT; i += 256) {
        const int r = i >> 6, c = i & 63;
        const float w = sEv[c];
        float f;
        if      (MODE == M_SQRT_BOTH || MODE == M_SQRT) f = sqrtf(fmaxf(w, 1e-30f));
        else if (MODE == M_LOG)     f = logf(fmaxf(w, 1e-30f));
        else if (MODE == M_EXP)     f = expf(w);
        else if (MODE == M_CLAMP)   f = fmaxf(w, 1e-4f);
        else /* M_INVSQRT */        f = 1.0f / sqrtf(fmaxf(w, 1e-30f));
        sA[r * LDSP + c] = sV[r * LDSP + c] * f;
    }
    __syncthreads();
    gemm64(sA, sV, /*transB=*/true, out1 + base, NMAT);

    if (MODE == M_SQRT_BOTH) {    // pass 2: inverse sqrt
        __syncthreads();
        for (int i = tid; i < NMAT * NMAT; i += 256) {
            const int r = i >> 6, c = i & 63;
            const float f = 1.0f / sqrtf(fmaxf(sEv[c], 1e-30f));
            sA[r * LDSP + c] = sV[r * LDSP + c] * f;
        }
        __syncthreads();
        gemm64(sA, sV, true, out2 + base, NMAT);
    }
}

// ---------------------------------------------------------------------------
// Congruence: out[b] = P @ X @ P   (P symmetric).
// pstride/xstride in floats (0 => broadcast). xmode==1: X indexed as pair
// channel of x[N,16,64,64] with chanoff in {0,2}.
// ---------------------------------------------------------------------------
__global__ __launch_bounds__(256)
void k_congr(const float* __restrict__ P, long pstride,
             const float* __restrict__ X, long xstride, int xmode, int chanoff,
             float* __restrict__ out)
{
    __shared__ float sP[NMAT * LDSP];
    __shared__ float sX[NMAT * LDSP];
    __shared__ float sT[NMAT * LDSP];

    const int tid = threadIdx.x;
    const int b   = blockIdx.x;

    const float* px;
    if (xmode == 1) {
        const int n = b >> 3, pr = b & 7;
        const int ch = ((pr >> 1) << 2) + (pr & 1) + chanoff;  // idx1 = {0,1,4,5,8,9,12,13}
        px = X + ((long)(n * 16 + ch)) * (NMAT * NMAT);
    } else {
        px = X + (long)b * xstride;
    }
    const float* pp = P + (long)b * pstride;

    for (int i = tid; i < NMAT * NMAT; i += 256) {
        const int r = i >> 6, c = i & 63;
        sP[r * LDSP + c] = pp[i];
        sX[r * LDSP + c] = px[i];
    }
    __syncthreads();
    gemm64(sP, sX, false, sT, LDSP);               // T = P @ X   (LDS)
    __syncthreads();
    gemm64(sT, sP, false, out + (long)b * (NMAT * NMAT), NMAT);  // T @ P
}

// ---------------------------------------------------------------------------
// Elementwise helpers
// ---------------------------------------------------------------------------
__device__ __forceinline__ void softmax2(const float* w1raw, float& w0, float& w1)
{
    const float a0 = w1raw[0], a1 = w1raw[1];
    const float m  = fmaxf(a0, a1);
    const float e0 = expf(a0 - m), e1 = expf(a1 - m);
    const float inv = 1.0f / (e0 + e1);
    w0 = e0 * inv; w1 = e1 * inv;
}

// G[b] = w0 * x[n, c] + w1 * x[n, c+2],  b = n*8 + pr
__global__ void k_pair_combine(const float* __restrict__ x,
                               const float* __restrict__ w1raw,
                               float* __restrict__ outg)
{
    float w0, w1; softmax2(w1raw, w0, w1);
    const long total = 4096l * 4096l;
    for (long i = (long)blockIdx.x * 256 + threadIdx.x; i < total;
         i += (long)gridDim.x * 256) {
        const long b = i >> 12;
        const int  e = (int)(i & 4095);
        const int  n = (int)(b >> 3), pr = (int)(b & 7);
        const int  c = ((pr >> 1) << 2) + (pr & 1);
        const long base = ((long)(n * 16 + c)) << 12;
        outg[i] = w0 * x[base + e] + w1 * x[base + 2 * 4096 + e];
    }
}

__global__ void k_axpy(const float* __restrict__ A, const float* __restrict__ B,
                       const float* __restrict__ w1raw, float* __restrict__ out,
                       long total)
{
    float w0, w1; softmax2(w1raw, w0, w1);
    for (long i = (long)blockIdx.x * 256 + threadIdx.x; i < total;
         i += (long)gridDim.x * 256)
        out[i] = w0 * A[i] + w1 * B[i];
}

// out[j] = mean_b in[b*4096 + j]
__global__ void k_mean(const float* __restrict__ in, float* __restrict__ out,
                       int batch)
{
    const int j = blockIdx.x * 256 + threadIdx.x;
    if (j < 4096) {
        float s = 0.0f;
        for (int b = 0; b < batch; ++b) s += in[(long)b * 4096 + j];
        out[j] = s * (1.0f / (float)batch);
    }
}

// ---------------------------------------------------------------------------
extern "C" void kernel_launch(void* const* d_in, const int* in_sizes, int n_in,
                              void* d_out, int out_size, void* d_ws, size_t ws_size,
                              hipStream_t stream)
{
    (void)in_sizes; (void)n_in; (void)out_size; (void)ws_size;
    const float* x   = (const float*)d_in[0];   // [512,16,64,64]
    const float* w1  = (const float*)d_in[1];   // [2]
    const float* bnw = (const float*)d_in[2];   // [64,64]
    float* out = (float*)d_out;                 // [512,8,64,64]
    float* ws  = (float*)d_ws;

    const int  B    = 4096;                 // 512 * 8 matrices
    const long MATS = (long)B * NMAT * NMAT;

    float* b0 = ws;
    float* b1 = ws + 1 * MATS;
    float* b2 = ws + 2 * MATS;
    float* b3 = ws + 3 * MATS;
    float* sm = ws + 4 * MATS;              // small single-matrix scratch
    float* s_gm   = sm + 0 * 4096;
    float* s_gms  = sm + 1 * 4096;
    float* s_gmis = sm + 2 * 4096;
    float* s_lm   = sm + 3 * 4096;
    float* s_le   = sm + 4 * 4096;
    float* s_g2   = sm + 5 * 4096;
    float* s_g2is = sm + 6 * 4096;
    float* s_wsq  = sm + 7 * 4096;

    const dim3 blk(256);

    // ---- stage A: weighted pair barycenter (one Karcher step) ----
    k_pair_combine<<<1024, blk, 0, stream>>>(x, w1, b0);                     // G
    k_eigh_fn<M_SQRT_BOTH><<<B, blk, 0, stream>>>(b0, b1, b2);               // Gs, Gis
    k_congr<<<B, blk, 0, stream>>>(b2, 4096, x, 0, 1, 0, b0);                // Gis Xa Gis
    k_congr<<<B, blk, 0, stream>>>(b2, 4096, x, 0, 1, 2, b3);                // Gis Xb Gis
    k_eigh_fn<M_LOG><<<B, blk, 0, stream>>>(b0, b0, nullptr);                // L1
    k_eigh_fn<M_LOG><<<B, blk, 0, stream>>>(b3, b3, nullptr);                // L2
    k_axpy<<<1024, blk, 0, stream>>>(b0, b3, w1, b2, MATS);                  // S
    k_eigh_fn<M_EXP><<<B, blk, 0, stream>>>(b2, b2, nullptr);                // expm(S)
    k_congr<<<B, blk, 0, stream>>>(b1, 4096, b2, 4096, 0, 0, b0);            // M = Gs E Gs

    // ---- stage B: SPD batch-norm (Karcher mean, 1 iter) + bias ----
    k_mean<<<16, blk, 0, stream>>>(b0, s_gm, B);                             // arith mean
    k_eigh_fn<M_SQRT_BOTH><<<1, blk, 0, stream>>>(s_gm, s_gms, s_gmis);
    k_congr<<<B, blk, 0, stream>>>(s_gmis, 0, b0, 4096, 0, 0, b1);           // whiten
    k_eigh_fn<M_LOG><<<B, blk, 0, stream>>>(b1, b1, nullptr);                // logs
    k_mean<<<16, blk, 0, stream>>>(b1, s_lm, B);                             // mean log
    k_eigh_fn<M_EXP><<<1, blk, 0, stream>>>(s_lm, s_le, nullptr);
    k_congr<<<1, blk, 0, stream>>>(s_gms, 0, s_le, 0, 0, 0, s_g2);           // G2 (Karcher)
    k_eigh_fn<M_INVSQRT><<<1, blk, 0, stream>>>(s_g2, s_g2is, nullptr);
    k_congr<<<B, blk, 0, stream>>>(s_g2is, 0, b0, 4096, 0, 0, b1);           // center
    k_eigh_fn<M_SQRT><<<1, blk, 0, stream>>>(bnw, s_wsq, nullptr);           // W^{1/2}
    k_congr<<<B, blk, 0, stream>>>(s_wsq, 0, b1, 4096, 0, 0, b0);            // bias

    // ---- ReEig: clamp eigenvalues, write output ----
    k_eigh_fn<M_CLAMP><<<B, blk, 0, stream>>>(b0, out, nullptr);
}